// Loss_71476845740753
// MI455X (gfx1250) — compile-verified
//
#include <hip/hip_runtime.h>
#include <math.h>

typedef float v2f __attribute__((ext_vector_type(2)));
typedef float v8f __attribute__((ext_vector_type(8)));
typedef int   v4i __attribute__((ext_vector_type(4)));

#define DIMK     512
#define EPS_PD   1e-6f
#define EPS_NORM 1e-12f
#define MARGIN   0.2f

#define BM 64
#define BN 128
#define BK 32
#define LDPAD (BK + 4)   // 36-float LDS row stride -> conflict-free b64 fragment reads

#if defined(__gfx1250__) && __has_builtin(__builtin_amdgcn_global_load_async_to_lds_b128) && __has_builtin(__builtin_amdgcn_s_wait_asynccnt)
#define USE_ASYNC_LDS 1
#else
#define USE_ASYNC_LDS 0
#endif

// ---------------------------------------------------------------------------
// Kernel A: n_t = n[0] / max(||n[0]||, eps)
// ---------------------------------------------------------------------------
__global__ void norm_neg_kernel(const float* __restrict__ n, float* __restrict__ nt) {
    __shared__ float red[256];
    __shared__ float s_inv;
    const int t = threadIdx.x;
    const float a = n[t];
    const float b = n[t + 256];
    red[t] = a * a + b * b;
    __syncthreads();
    for (int s = 128; s > 0; s >>= 1) {
        if (t < s) red[t] += red[t + s];
        __syncthreads();
    }
    if (t == 0) s_inv = 1.0f / fmaxf(sqrtf(red[0]), EPS_NORM);
    __syncthreads();
    nt[t]       = a * s_inv;
    nt[t + 256] = b * s_inv;
}

// ---------------------------------------------------------------------------
// Kernel B: per-row stats of p
// ---------------------------------------------------------------------------
__global__ void row_stats_kernel(const float* __restrict__ p,
                                 const float* __restrict__ nt,
                                 float* __restrict__ inv,
                                 float* __restrict__ xx,
                                 float* __restrict__ sx,
                                 float* __restrict__ dpn) {
    __shared__ float r1[256];
    __shared__ float r2[256];
    __shared__ float s_inv, s_ss, s_sum;
    const int row = blockIdx.x;
    const int t   = threadIdx.x;
    const float* pr = p + (size_t)row * DIMK;
    const float a = pr[t];
    const float b = pr[t + 256];
    r1[t] = a * a + b * b;
    r2[t] = a + b;
    __syncthreads();
    for (int s = 128; s > 0; s >>= 1) {
        if (t < s) { r1[t] += r1[t + s]; r2[t] += r2[t + s]; }
        __syncthreads();
    }
    if (t == 0) {
        s_ss  = r1[0];
        s_sum = r2[0];
        s_inv = 1.0f / fmaxf(sqrtf(r1[0]), EPS_NORM);
    }
    __syncthreads();
    const float iv = s_inv;
    const float d0 = a * iv - nt[t] + EPS_PD;
    const float d1 = b * iv - nt[t + 256] + EPS_PD;
    __syncthreads();
    r1[t] = d0 * d0 + d1 * d1;
    __syncthreads();
    for (int s = 128; s > 0; s >>= 1) {
        if (t < s) r1[t] += r1[t + s];
        __syncthreads();
    }
    if (t == 0) {
        inv[row] = iv;
        xx[row]  = s_ss * iv * iv;
        sx[row]  = s_sum * iv;
        dpn[row] = sqrtf(r1[0]);
    }
}

// ---------------------------------------------------------------------------
// Kernel C: fused Gram GEMM (fp32 WMMA 16x16x4) + distance/hinge epilogue.
// 256 threads = 8 waves (2 waveM x 4 waveN); each wave computes a 32x32
// output patch as 2x2 WMMA tiles (4 independent accumulator chains).
// Block tile 64(M) x 128(N); K staged in BK=32 chunks, double-buffered LDS
// filled by GLOBAL_LOAD_ASYNC_TO_LDS_B128 (ASYNCcnt) when available.
// ---------------------------------------------------------------------------
__global__ __launch_bounds__(256) void gram_loss_kernel(
        const float* __restrict__ p,
        const float* __restrict__ inv,
        const float* __restrict__ xx,
        const float* __restrict__ sx,
        const float* __restrict__ dpn,
        float* __restrict__ partials) {
    __shared__ float As[2][BM][LDPAD];
    __shared__ float Bs[2][BN][LDPAD];
    __shared__ float red[256];

    const int t     = threadIdx.x;
    const int lane  = t & 31;
    const int wave  = t >> 5;
    const int waveM = wave >> 2;   // 0..1
    const int waveN = wave & 3;    // 0..3
    const int lo    = lane & 15;   // 0..15
    const int hi    = lane >> 4;   // 0..1

    const int iblk = blockIdx.y * BM;
    const int jblk = blockIdx.x * BN;

    // ---- staging: one BK-chunk of A (64xBK) and B (128xBK) into LDS buffer buf ----
    auto stage = [&](int kc, int buf) {
        // A: 64*32 floats = 512 float4 slots, 2 per thread (8 float4 per row)
#pragma unroll
        for (int q = 0; q < 2; ++q) {
            const int s  = q * 256 + t;
            const int r  = s >> 3;
            const int c4 = (s & 7) << 2;
            const float* src = p + (size_t)(iblk + r) * DIMK + kc + c4;
#if USE_ASYNC_LDS
            __builtin_amdgcn_global_load_async_to_lds_b128(
                (__attribute__((address_space(1))) v4i*)src,
                (__attribute__((address_space(3))) v4i*)&As[buf][r][c4], 0, 0);
#else
            const float4 v = *(const float4*)src;
            As[buf][r][c4 + 0] = v.x; As[buf][r][c4 + 1] = v.y;
            As[buf][r][c4 + 2] = v.z; As[buf][r][c4 + 3] = v.w;
#endif
        }
        // B: 128*32 floats = 1024 float4 slots, 4 per thread
#pragma unroll
        for (int q = 0; q < 4; ++q) {
            const int s  = q * 256 + t;
            const int r  = s >> 3;
            const int c4 = (s & 7) << 2;
            const float* src = p + (size_t)(jblk + r) * DIMK + kc + c4;
#if USE_ASYNC_LDS
            __builtin_amdgcn_global_load_async_to_lds_b128(
                (__attribute__((address_space(1))) v4i*)src,
                (__attribute__((address_space(3))) v4i*)&Bs[buf][r][c4], 0, 0);
#else
            const float4 v = *(const float4*)src;
            Bs[buf][r][c4 + 0] = v.x; Bs[buf][r][c4 + 1] = v.y;
            Bs[buf][r][c4 + 2] = v.z; Bs[buf][r][c4 + 3] = v.w;
#endif
        }
    };

    v8f acc00 = {}, acc01 = {}, acc10 = {}, acc11 = {};

    stage(0, 0);
    int buf = 0;
    for (int kc = 0; kc < DIMK; kc += BK) {
#if USE_ASYNC_LDS
        __builtin_amdgcn_s_wait_asynccnt(0);
#endif
        __syncthreads();   // LDS buffer buf complete for all waves; prior reads of buf^1 done

        if (kc + BK < DIMK) stage(kc + BK, buf ^ 1);   // overlaps with compute below

        // Fragment layout (V_WMMA_F32_16X16X4_F32):
        //   A: lane -> M = lane%16, K = 2*(lane/16) + v
        //   B: lane -> N = lane%16, K = 2*(lane/16) + v
        const float* arow0 = &As[buf][waveM * 32 + lo][0];
        const float* arow1 = arow0 + 16 * LDPAD;
        const float* brow0 = &Bs[buf][waveN * 32 + lo][0];
        const float* brow1 = brow0 + 16 * LDPAD;
#pragma unroll
        for (int k = 0; k < BK; k += 4) {
            const v2f a0 = *(const v2f*)(arow0 + k + 2 * hi);
            const v2f a1 = *(const v2f*)(arow1 + k + 2 * hi);
            const v2f b0 = *(const v2f*)(brow0 + k + 2 * hi);
            const v2f b1 = *(const v2f*)(brow1 + k + 2 * hi);
            acc00 = __builtin_amdgcn_wmma_f32_16x16x4_f32(false, a0, false, b0, (short)0, acc00, false, false);
            acc01 = __builtin_amdgcn_wmma_f32_16x16x4_f32(false, a0, false, b1, (short)0, acc01, false, false);
            acc10 = __builtin_amdgcn_wmma_f32_16x16x4_f32(false, a1, false, b0, (short)0, acc10, false, false);
            acc11 = __builtin_amdgcn_wmma_f32_16x16x4_f32(false, a1, false, b1, (short)0, acc11, false, false);
        }
        buf ^= 1;
    }

    // Epilogue: C layout -> VGPR r, lane: M = r + 8*(lane/16), N = lane%16
    const int ibase = iblk + waveM * 32 + 8 * hi;
    const int jbase = jblk + waveN * 32 + lo;
    const float eps2d = (EPS_PD * EPS_PD) * (float)DIMK;

    float lsum = 0.0f;
#pragma unroll
    for (int tn = 0; tn < 2; ++tn) {
        const int   j    = jbase + tn * 16;
        const float invj = inv[j];
        const float xxj  = xx[j];
        const float sxj  = sx[j];
#pragma unroll
        for (int tm = 0; tm < 2; ++tm) {
            const v8f accv = (tm == 0) ? (tn == 0 ? acc00 : acc01)
                                       : (tn == 0 ? acc10 : acc11);
#pragma unroll
            for (int r = 0; r < 8; ++r) {
                const int   i  = ibase + tm * 16 + r;
                const float g  = accv[r] * inv[i] * invj;
                float d2 = xx[i] + xxj - 2.0f * g
                         + 2.0f * EPS_PD * (sx[i] - sxj) + eps2d;
                const float dpp = sqrtf(fmaxf(d2, 1e-12f));
                lsum += fmaxf(0.0f, dpp + MARGIN - dpn[i]);
            }
        }
    }

    red[t] = lsum;
    __syncthreads();
    for (int s = 128; s > 0; s >>= 1) {
        if (t < s) red[t] += red[t + s];
        __syncthreads();
    }
    if (t == 0) partials[blockIdx.y * gridDim.x + blockIdx.x] = red[0];
}

// ---------------------------------------------------------------------------
// Kernel D: deterministic final reduction (double accumulation)
// ---------------------------------------------------------------------------
__global__ void final_reduce_kernel(const float* __restrict__ partials, int np,
                                    float* __restrict__ out, double denom) {
    __shared__ double red[256];
    const int t = threadIdx.x;
    double s = 0.0;
    for (int i = t; i < np; i += 256) s += (double)partials[i];
    red[t] = s;
    __syncthreads();
    for (int st = 128; st > 0; st >>= 1) {
        if (t < st) red[t] += red[t + st];
        __syncthreads();
    }
    if (t == 0) {
        const double loss = red[0] / denom;
        out[0] = (float)fmax(loss, 0.0);
    }
}

// ---------------------------------------------------------------------------
extern "C" void kernel_launch(void* const* d_in, const int* in_sizes, int n_in,
                              void* d_out, int out_size, void* d_ws, size_t ws_size,
                              hipStream_t stream) {
    const float* p = (const float*)d_in[0];
    const float* n = (const float*)d_in[1];
    const int LP = in_sizes[0] / DIMK;   // 8192

    float* ws       = (float*)d_ws;
    float* nt       = ws;                 // [DIMK]
    float* inv      = nt + DIMK;          // [LP]
    float* xx       = inv + LP;           // [LP]
    float* sx       = xx + LP;            // [LP]
    float* dpn      = sx + LP;            // [LP]
    float* partials = dpn + LP;           // [gx*gy]

    const int gx = LP / BN;               // 64
    const int gy = LP / BM;               // 128

    norm_neg_kernel<<<1, 256, 0, stream>>>(n, nt);
    row_stats_kernel<<<LP, 256, 0, stream>>>(p, nt, inv, xx, sx, dpn);
    gram_loss_kernel<<<dim3(gx, gy), 256, 0, stream>>>(p, inv, xx, sx, dpn, partials);
    final_reduce_kernel<<<1, 256, 0, stream>>>(partials, gx * gy, (float*)d_out,
                                               (double)LP * (double)(LP - 1));
}